// GraphSAGE_21887153340604
// MI455X (gfx1250) — compile-verified
//
#include <hip/hip_runtime.h>

typedef float v2f __attribute__((ext_vector_type(2)));
typedef float v4f __attribute__((ext_vector_type(4)));
typedef float v8f __attribute__((ext_vector_type(8)));

#define NN 100000
#define DIN 128
#define DH  128
#define DOUT 64
#define LDS_STRIDE 132   // 132 = 4*33: 16B-aligned rows, 4r+k bank pattern -> no conflicts

__device__ __forceinline__ v8f wmma_f32_k4(v2f a, v2f b, v8f c) {
    // V_WMMA_F32_16X16X4_F32 : D(16x16 f32) = A(16x4 f32) * B(4x16 f32) + C
    return __builtin_amdgcn_wmma_f32_16x16x4_f32(
        false, a, false, b, (short)0, c, false, false);
}

// ---------------- degree counts ----------------
__global__ void deg_kernel(const int* __restrict__ er, float* __restrict__ deg, int E) {
    int e = blockIdx.x * blockDim.x + threadIdx.x;
    if (e < E) atomicAdd(&deg[er[e]], 1.0f);
}

// ---------------- pack W into WMMA-B fragment order ----------------
// wp laid out so that iteration (kk, colTile) is one contiguous 256B block:
//   wp_v2f[(kk*nct + ct)*32 + lane] = ( W[k0][col], W[k0+1][col] )
//   k0 = 4*kk + 2*(lane>=16), col = ct*16 + (lane&15)
__global__ void pack_kernel(const float* __restrict__ W, float* __restrict__ wp,
                            int ncols, int nct) {
    int tid = blockIdx.x * blockDim.x + threadIdx.x;
    int total = 32 * nct * 32;              // (K/4) steps * colTiles * 32 lanes
    if (tid < total) {
        int lane = tid & 31;
        int ct   = (tid >> 5) % nct;
        int kk   = tid / (32 * nct);
        int k0   = 4 * kk + ((lane >> 4) << 1);
        int col  = ct * 16 + (lane & 15);
        wp[2 * tid]     = W[(size_t)k0 * ncols + col];
        wp[2 * tid + 1] = W[(size_t)(k0 + 1) * ncols + col];
    }
}

// ---------------- GEMM1: h0 = x @ W1  [N x 128] ----------------
__global__ __launch_bounds__(256) void gemm1_kernel(const float* __restrict__ x,
                                                    const float* __restrict__ wp1,
                                                    float* __restrict__ h0) {
    __shared__ float xs[16 * LDS_STRIDE];
    const int t = threadIdx.x;
    const int rowBase = blockIdx.x * 16;
    // stage A tile 16x128 into LDS as float4 (b128 load + b128 ds_store)
#pragma unroll
    for (int i = 0; i < 2; ++i) {
        int idx4 = t + 256 * i;             // 0..511 float4 slots
        int r = idx4 >> 5, kq = idx4 & 31;
        v4f v = *(const v4f*)(x + (size_t)(rowBase + r) * DIN + kq * 4);
        *(v4f*)(xs + r * LDS_STRIDE + kq * 4) = v;
    }
    __syncthreads();

    const int wave = t >> 5, lane = t & 31;
    const int col0 = wave * 16;             // 8 waves cover 128 output columns
    const int arow = lane & 15;
    const int khalf = (lane >> 4) << 1;     // lanes 16..31 hold K+2,K+3
    const v2f* Wp = (const v2f*)wp1;

    v8f c = {};
#pragma unroll
    for (int kk = 0; kk < DIN / 4; ++kk) {
        int k0 = kk * 4 + khalf;
        v2f a, b;
        a.x = xs[arow * LDS_STRIDE + k0];
        a.y = xs[arow * LDS_STRIDE + k0 + 1];
        b = Wp[(kk * 8 + wave) * 32 + lane];        // coalesced b64
        c = wmma_f32_k4(a, b, c);
    }
    const int srow = (lane >> 4) * 8;       // D: VGPR r -> rows r (lanes<16) / r+8
#pragma unroll
    for (int r = 0; r < 8; ++r) {
        int m = r + srow;
        h0[(size_t)(rowBase + m) * DH + col0 + arow] = c[r];
    }
}

// ---------------- edge scatter-add: acc[row] += h[col], D channels ----------------
// one thread per (edge, 4-channel group): 1 index pair, 1 b128 gather, 4 f32 atomics
template <int D>
__global__ void agg_kernel(const float* __restrict__ h, const int* __restrict__ er,
                           const int* __restrict__ ec, float* __restrict__ acc, int E) {
    constexpr int G = D / 4;
    unsigned tid = blockIdx.x * blockDim.x + threadIdx.x;
    unsigned e = tid / G;
    unsigned g = tid % G;
    if (e < (unsigned)E) {
        v4f hv = *(const v4f*)(h + (size_t)ec[e] * D + g * 4);
        float* a = acc + (size_t)er[e] * D + g * 4;
        atomicAdd(a + 0, hv.x);
        atomicAdd(a + 1, hv.y);
        atomicAdd(a + 2, hv.z);
        atomicAdd(a + 3, hv.w);
    }
}

// ---------------- GEMM2: h2 = relu(acc1/deg) @ W2  [N x 64], relu/div fused ----------------
__global__ __launch_bounds__(256) void gemm2_kernel(const float* __restrict__ acc1,
                                                    const float* __restrict__ deg,
                                                    const float* __restrict__ wp2,
                                                    float* __restrict__ h2) {
    __shared__ float hs[32 * LDS_STRIDE];
    const int t = threadIdx.x;
    const int rowBase = blockIdx.x * 32;
    // stage A tile 32x128 with fused mean + relu, float4-wide
#pragma unroll
    for (int i = 0; i < 4; ++i) {
        int idx4 = t + 256 * i;             // 0..1023 float4 slots
        int r = idx4 >> 5, kq = idx4 & 31;
        int gg = rowBase + r;
        float d = deg[gg];
        v4f v = *(const v4f*)(acc1 + (size_t)gg * DH + kq * 4);
        v4f o;
        o.x = fmaxf(v.x / d, 0.0f);
        o.y = fmaxf(v.y / d, 0.0f);
        o.z = fmaxf(v.z / d, 0.0f);
        o.w = fmaxf(v.w / d, 0.0f);
        *(v4f*)(hs + r * LDS_STRIDE + kq * 4) = o;
    }
    __syncthreads();

    const int wave = t >> 5, lane = t & 31;
    const int rtile = (wave >> 2) * 16;     // 2 row tiles
    const int ct    = wave & 3;             // 4 col tiles cover 64 outputs
    const int col0  = ct * 16;
    const int arow = lane & 15;
    const int khalf = (lane >> 4) << 1;
    const v2f* Wp = (const v2f*)wp2;

    v8f c = {};
#pragma unroll
    for (int kk = 0; kk < DH / 4; ++kk) {
        int k0 = kk * 4 + khalf;
        v2f a, b;
        a.x = hs[(rtile + arow) * LDS_STRIDE + k0];
        a.y = hs[(rtile + arow) * LDS_STRIDE + k0 + 1];
        b = Wp[(kk * 4 + ct) * 32 + lane];          // coalesced b64
        c = wmma_f32_k4(a, b, c);
    }
    const int srow = (lane >> 4) * 8;
#pragma unroll
    for (int r = 0; r < 8; ++r) {
        int m = rtile + r + srow;
        h2[(size_t)(rowBase + m) * DOUT + col0 + arow] = c[r];
    }
}

// ---------------- final: out = acc2 / deg (float4-wide) ----------------
__global__ void final_kernel(const float* __restrict__ acc2, const float* __restrict__ deg,
                             float* __restrict__ out) {
    int tid = blockIdx.x * blockDim.x + threadIdx.x;   // NN*16 groups of 4
    int i = tid >> 4;
    if (i < NN) {
        v4f v = *(const v4f*)(acc2 + (size_t)tid * 4);
        float d = deg[i];
        v4f o;
        o.x = v.x / d; o.y = v.y / d; o.z = v.z / d; o.w = v.w / d;
        *(v4f*)(out + (size_t)tid * 4) = o;
    }
}

extern "C" void kernel_launch(void* const* d_in, const int* in_sizes, int n_in,
                              void* d_out, int out_size, void* d_ws, size_t ws_size,
                              hipStream_t stream) {
    const float* x  = (const float*)d_in[0];
    const float* W1 = (const float*)d_in[1];
    const float* W2 = (const float*)d_in[2];
    const int* er   = (const int*)d_in[3];
    const int* ec   = (const int*)d_in[4];
    float* out = (float*)d_out;
    const int E = in_sizes[3];

    // workspace layout (floats): deg | buf0 (h0 then h2) | buf1 (acc1 then acc2) | wp1 | wp2
    float* ws   = (float*)d_ws;
    float* deg  = ws;
    float* buf0 = ws + 100352;                       // N*128 floats
    float* buf1 = buf0 + (size_t)NN * DH;            // N*128 floats
    float* wp1  = buf1 + (size_t)NN * DH;            // 128*128 floats packed
    float* wp2  = wp1 + (size_t)DIN * DH;            // 128*64 floats packed

    hipMemsetAsync(deg, 0, (size_t)NN * sizeof(float), stream);
    hipMemsetAsync(buf1, 0, (size_t)NN * DH * sizeof(float), stream);

    pack_kernel<<<(32 * 8 * 32 + 255) / 256, 256, 0, stream>>>(W1, wp1, DH, 8);
    pack_kernel<<<(32 * 4 * 32 + 255) / 256, 256, 0, stream>>>(W2, wp2, DOUT, 4);

    deg_kernel<<<(E + 255) / 256, 256, 0, stream>>>(er, deg, E);
    gemm1_kernel<<<NN / 16, 256, 0, stream>>>(x, wp1, buf0);                // h0
    {
        unsigned total = (unsigned)E * (DH / 4);
        agg_kernel<DH><<<(total + 255) / 256, 256, 0, stream>>>(buf0, er, ec, buf1, E); // acc1
    }
    gemm2_kernel<<<NN / 32, 256, 0, stream>>>(buf1, deg, wp2, buf0);        // h2 (reuse buf0)
    hipMemsetAsync(buf1, 0, (size_t)NN * DOUT * sizeof(float), stream);     // acc2 (reuse buf1)
    {
        unsigned total = (unsigned)E * (DOUT / 4);
        agg_kernel<DOUT><<<(total + 255) / 256, 256, 0, stream>>>(buf0, er, ec, buf1, E);
    }
    final_kernel<<<(NN * 16 + 255) / 256, 256, 0, stream>>>(buf1, deg, out);
}